// HierarchicalLoss_8272107012591
// MI455X (gfx1250) — compile-verified
//
#include <hip/hip_runtime.h>

#define NB 1024          // partial-reduction blocks
#define NT 256           // threads per block (8 wave32s)
#define NUM_CLASSES 6

typedef __attribute__((ext_vector_type(2))) float v2f;
typedef __attribute__((ext_vector_type(8))) float v8f;

__device__ __forceinline__ float sel6(int t, float a0, float a1, float a2,
                                      float a3, float a4, float a5) {
    float r = a0;
    r = (t == 1) ? a1 : r;
    r = (t == 2) ? a2 : r;
    r = (t == 3) ? a3 : r;
    r = (t == 4) ? a4 : r;
    r = (t == 5) ? a5 : r;
    return r;
}

__device__ __forceinline__ float wave_reduce(float v) {
    #pragma unroll
    for (int off = 16; off > 0; off >>= 1)
        v += __shfl_xor(v, off, 32);
    return v;
}

// Pass 1: streaming softmax + three partial sums per block.
__global__ void __launch_bounds__(NT)
hloss_partial(const float* __restrict__ preds,
              const int* __restrict__ targets,
              const float* __restrict__ cw,
              const float* __restrict__ pen,
              float* __restrict__ ws, int B) {
    __shared__ float sP[36];
    __shared__ float sW[6];
    __shared__ float sRed[3][NT / 32];

    const int tid = threadIdx.x;
    if (tid < 36) sP[tid] = pen[tid];
    if (tid < 6)  sW[tid] = cw[tid];
    __syncthreads();

    float s1 = 0.f, s2 = 0.f, s3 = 0.f;
    const int stride = NB * NT;
    for (int i = blockIdx.x * NT + tid; i < B; i += stride) {
        const float* p = preds + (size_t)i * NUM_CLASSES;
        // Speculative prefetch one grid-stride ahead (gfx1250 global_prefetch;
        // OOB speculative prefetches are silently dropped).
        __builtin_prefetch(p + (size_t)stride * NUM_CLASSES, 0, 1);

        // Row stride is 24B -> three 8B-aligned b64 loads (b128 would misalign).
        v2f va = *(const v2f*)(p);
        v2f vb = *(const v2f*)(p + 2);
        v2f vc = *(const v2f*)(p + 4);
        float p0 = va.x, p1 = va.y, p2 = vb.x, p3 = vb.y, p4 = vc.x, p5 = vc.y;
        int t = targets[i];

        float m  = fmaxf(fmaxf(fmaxf(p0, p1), fmaxf(p2, p3)), fmaxf(p4, p5));
        float e0 = __expf(p0 - m), e1 = __expf(p1 - m), e2 = __expf(p2 - m);
        float e3 = __expf(p3 - m), e4 = __expf(p4 - m), e5 = __expf(p5 - m);
        float se  = ((e0 + e1) + (e2 + e3)) + (e4 + e5);
        float inv = __builtin_amdgcn_rcpf(se);
        float lse = __logf(se) + m;

        float pt  = sel6(t, p0, p1, p2, p3, p4, p5);
        float et  = sel6(t, e0, e1, e2, e3, e4, e5);
        float nll = lse - pt;

        const float* pr = sP + t * NUM_CLASSES;     // LDS gather of penalty row
        float q0 = pr[0], q1 = pr[1], q2 = pr[2], q3 = pr[3], q4 = pr[4], q5 = pr[5];
        float hd  = fmaf(e0, q0, fmaf(e1, q1, fmaf(e2, q2,
                    fmaf(e3, q3, fmaf(e4, q4, e5 * q5)))));
        float qtt = sel6(t, q0, q1, q2, q3, q4, q5); // pen[t,t]
        hd -= et * qtt;                              // drop j == t term

        float wt = sW[t];
        s1 = fmaf(wt, nll, s1);
        s2 += wt;
        s3 = fmaf(hd, inv, s3);
    }

    s1 = wave_reduce(s1);
    s2 = wave_reduce(s2);
    s3 = wave_reduce(s3);
    const int lane = tid & 31, wid = tid >> 5;
    if (lane == 0) { sRed[0][wid] = s1; sRed[1][wid] = s2; sRed[2][wid] = s3; }
    __syncthreads();
    if (tid == 0) {
        float r1 = 0.f, r2 = 0.f, r3 = 0.f;
        #pragma unroll
        for (int w = 0; w < NT / 32; ++w) {
            r1 += sRed[0][w]; r2 += sRed[1][w]; r3 += sRed[2][w];
        }
        ws[blockIdx.x]          = r1;
        ws[NB + blockIdx.x]     = r2;
        ws[2 * NB + blockIdx.x] = r3;
    }
}

// Pass 2 (one wave32, EXEC all ones): matrix-engine reduction of the block
// partials via V_WMMA_F32_16X16X4_F32 with A == all-ones. With A==1,
// D[m,n] = sum_k B[k,n], so the sum of ALL 256 D elements equals
// 16 * (sum of all 64 B elements) independent of lane/VGPR striping.
__global__ void __launch_bounds__(32)
hloss_finalize(const float* __restrict__ ws, float* __restrict__ out,
               float invB, float hw) {
    const int lane = threadIdx.x;
    v2f ones; ones.x = 1.0f; ones.y = 1.0f;
    float sums[3];
    #pragma unroll
    for (int a = 0; a < 3; ++a) {
        const float* base = ws + a * NB;
        v8f acc = {};
        for (int i = 0; i < NB; i += 64) {   // NB % 64 == 0, no tail
            v2f bm;
            bm.x = base[i + lane];
            bm.y = base[i + 32 + lane];
            acc = __builtin_amdgcn_wmma_f32_16x16x4_f32(
                false, ones, false, bm, (short)0, acc, false, false);
        }
        float loc = ((acc[0] + acc[1]) + (acc[2] + acc[3])) +
                    ((acc[4] + acc[5]) + (acc[6] + acc[7]));
        loc = wave_reduce(loc);
        sums[a] = loc * (1.0f / 16.0f);
    }
    if (lane == 0) {
        float ce   = sums[0] / sums[1];
        float hier = sums[2] * invB;
        out[0] = fmaf(hw, hier, ce);  // total
        out[1] = ce;
        out[2] = hier;
    }
}

extern "C" void kernel_launch(void* const* d_in, const int* in_sizes, int n_in,
                              void* d_out, int out_size, void* d_ws, size_t ws_size,
                              hipStream_t stream) {
    const float* preds   = (const float*)d_in[0];
    const int*   targets = (const int*)d_in[1];
    const float* cw      = (const float*)d_in[2];
    const float* pen     = (const float*)d_in[3];
    float* out = (float*)d_out;
    float* ws  = (float*)d_ws;   // needs 3*NB*4 = 12 KB
    const int B = in_sizes[0] / NUM_CLASSES;

    hipLaunchKernelGGL(hloss_partial, dim3(NB), dim3(NT), 0, stream,
                       preds, targets, cw, pen, ws, B);
    hipLaunchKernelGGL(hloss_finalize, dim3(1), dim3(32), 0, stream,
                       ws, out, 1.0f / (float)B, 0.2f);
}